// GCN_train_56040733278666
// MI455X (gfx1250) — compile-verified
//
#include <hip/hip_runtime.h>
#include <math.h>

typedef float v2f __attribute__((ext_vector_type(2)));
typedef float v8f __attribute__((ext_vector_type(8)));

#define N_NODES   100000
#define N_EDGES   3200000
#define N_GRAPHS  64
#define IN_FEATS  128
#define DIM       32

// ---------------------------------------------------------------------------
// Fused node GEMM on the FP32 WMMA pipe (V_WMMA_F32_16X16X4_F32).
// One wave computes a 16-row x 32-col tile of BOTH products:
//   rhs   = H @ Wn                (feeds the edge scatter)
//   neigh = H @ Ws + bn           (self term + bias, seeds the accumulator)
// A layout (16x4 f32): lane<16 -> K={kk,kk+1}, lane>=16 -> K={kk+2,kk+3}, M=lane%16
// B layout (4x16 f32): same K split, lane%16 indexes N
// D layout: VGPR r -> M = r + 8*(lane/16), N = lane%16
// ---------------------------------------------------------------------------
__global__ void gemm_dual_wmma(const float* __restrict__ H, int ldh, int K,
                               const float* __restrict__ Wn,
                               const float* __restrict__ Ws,
                               const float* __restrict__ bn,
                               float* __restrict__ rhs,
                               float* __restrict__ neigh,
                               int nNodes)
{
    const int lane = threadIdx.x & 31;
    const int wave = threadIdx.x >> 5;
    const int wavesPerBlock = blockDim.x >> 5;
    const int tile = blockIdx.x * wavesPerBlock + wave;   // wave-uniform
    const int row0 = tile * 16;
    if (row0 >= nNodes) return;                           // whole wave exits together

    const int half = lane >> 4;        // 0 or 1
    const int tm   = lane & 15;
    const int arow = row0 + tm;        // A-matrix row this lane feeds

    v8f accN0 = {}; v8f accN1 = {};    // H@Wn, col tiles [0..15], [16..31]
    v8f accS0 = {}; v8f accS1 = {};    // H@Ws

    const float* hrow = H + (size_t)arow * (size_t)ldh;

    for (int kk = 0; kk < K; kk += 4) {
        const int ka = kk + 2 * half;
        v2f a;
        a.x = hrow[ka];
        a.y = hrow[ka + 1];

        const float* wn = Wn + (size_t)ka * DIM + tm;
        const float* ws = Ws + (size_t)ka * DIM + tm;
        v2f bN0; bN0.x = wn[0];        bN0.y = wn[DIM];
        v2f bN1; bN1.x = wn[16];       bN1.y = wn[DIM + 16];
        v2f bS0; bS0.x = ws[0];        bS0.y = ws[DIM];
        v2f bS1; bS1.x = ws[16];       bS1.y = ws[DIM + 16];

        accN0 = __builtin_amdgcn_wmma_f32_16x16x4_f32(false, a, false, bN0,
                                                      (short)0, accN0, false, false);
        accN1 = __builtin_amdgcn_wmma_f32_16x16x4_f32(false, a, false, bN1,
                                                      (short)0, accN1, false, false);
        accS0 = __builtin_amdgcn_wmma_f32_16x16x4_f32(false, a, false, bS0,
                                                      (short)0, accS0, false, false);
        accS1 = __builtin_amdgcn_wmma_f32_16x16x4_f32(false, a, false, bS1,
                                                      (short)0, accS1, false, false);
    }

    const float b0 = bn[tm];
    const float b1 = bn[16 + tm];
#pragma unroll
    for (int r = 0; r < 8; ++r) {
        const int m = row0 + r + 8 * half;
        rhs  [(size_t)m * DIM + tm]      = accN0[r];
        rhs  [(size_t)m * DIM + 16 + tm] = accN1[r];
        neigh[(size_t)m * DIM + tm]      = accS0[r] + b0;
        neigh[(size_t)m * DIM + 16 + tm] = accS1[r] + b1;
    }
}

// ---------------------------------------------------------------------------
// Edge scatter: one wave per edge, lane = feature column.
// 128 B coalesced gather from rhs[src], 128 B of contiguous f32 atomics
// into neigh[dst]; both arrays are 12.8 MB -> L2 resident.
// ---------------------------------------------------------------------------
__global__ void edge_scatter(const int* __restrict__ src, const int* __restrict__ dst,
                             const float* __restrict__ rhs, float* __restrict__ neigh)
{
    const int tid = blockIdx.x * blockDim.x + threadIdx.x;   // < 2^31
    const int e = tid >> 5;
    const int c = tid & 31;
    if (e >= N_EDGES) return;
    const int s = src[e];            // wave-uniform broadcast load
    const int d = dst[e];
    atomicAdd(neigh + (size_t)d * DIM + c, rhs[(size_t)s * DIM + c]);
}

__global__ void relu_copy(const float* __restrict__ in, float* __restrict__ out, int n)
{
    const int i = blockIdx.x * blockDim.x + threadIdx.x;
    if (i < n) {
        const float v = in[i];
        out[i] = v > 0.0f ? v : 0.0f;
    }
}

__global__ void zero_buf(float* __restrict__ p, int n)
{
    const int i = blockIdx.x * blockDim.x + threadIdx.x;
    if (i < n) p[i] = 0.0f;
}

// Per-graph sum pooling: wave per node, lane = column, atomics into 64x32.
__global__ void pool_sum(const float* __restrict__ h, const int* __restrict__ gid,
                         float* __restrict__ hg)
{
    const int tid = blockIdx.x * blockDim.x + threadIdx.x;
    const int n = tid >> 5;
    const int c = tid & 31;
    if (n >= N_NODES) return;
    const int g = gid[n];
    atomicAdd(hg + (size_t)g * DIM + c, h[(size_t)n * DIM + c]);
}

// Head: hg2 = relu(hg@Wfc1 + bfc1); o = relu(hg2@Wout + bout); softmax rows.
// 64 graphs -> one thread each; trivially small.
__global__ void head_kernel(const float* __restrict__ hg,
                            const float* __restrict__ Wfc1, const float* __restrict__ bfc1,
                            const float* __restrict__ Wout, const float* __restrict__ bout,
                            float* __restrict__ out)
{
    const int g = threadIdx.x;
    if (g >= N_GRAPHS) return;
    const float* r = hg + (size_t)g * DIM;

    float h2[8];
#pragma unroll
    for (int j = 0; j < 8; ++j) {
        float s = bfc1[j];
        for (int c = 0; c < DIM; ++c) s += r[c] * Wfc1[c * 8 + j];
        h2[j] = fmaxf(s, 0.0f);
    }
    float o[4];
#pragma unroll
    for (int j = 0; j < 4; ++j) {
        float s = bout[j];
        for (int c = 0; c < 8; ++c) s += h2[c] * Wout[c * 4 + j];
        o[j] = fmaxf(s, 0.0f);
    }
    float m = fmaxf(fmaxf(o[0], o[1]), fmaxf(o[2], o[3]));
    float e[4]; float sum = 0.0f;
#pragma unroll
    for (int j = 0; j < 4; ++j) { e[j] = expf(o[j] - m); sum += e[j]; }
    const float inv = 1.0f / sum;
#pragma unroll
    for (int j = 0; j < 4; ++j) out[(size_t)g * 4 + j] = e[j] * inv;
}

extern "C" void kernel_launch(void* const* d_in, const int* in_sizes, int n_in,
                              void* d_out, int out_size, void* d_ws, size_t ws_size,
                              hipStream_t stream) {
    (void)in_sizes; (void)n_in; (void)out_size; (void)ws_size;

    const float* x    = (const float*)d_in[0];   // [100000,128]
    const float* Wn0  = (const float*)d_in[1];   // [128,32]
    const float* bn0  = (const float*)d_in[2];   // [32]
    const float* Ws0  = (const float*)d_in[3];   // [128,32]
    const float* Wn   = (const float*)d_in[4];   // [3,32,32]
    const float* bn   = (const float*)d_in[5];   // [3,32]
    const float* Ws   = (const float*)d_in[6];   // [3,32,32]
    const float* Wfc1 = (const float*)d_in[7];   // [32,8]
    const float* bfc1 = (const float*)d_in[8];   // [8]
    const float* Wout = (const float*)d_in[9];   // [8,4]
    const float* bout = (const float*)d_in[10];  // [4]
    const int*   src  = (const int*)d_in[11];    // [3.2M]
    const int*   dst  = (const int*)d_in[12];    // [3.2M]
    const int*   gid  = (const int*)d_in[13];    // [100000]
    float* out = (float*)d_out;

    // Workspace layout (floats)
    const size_t NF = (size_t)N_NODES * DIM;     // 3.2M floats
    float* ws    = (float*)d_ws;
    float* rhs   = ws;                           // H @ Wn
    float* neigh = rhs + NF;                     // self + bias, then += scatter
    float* hA    = neigh + NF;                   // ping
    float* hB    = hA + NF;                      // pong
    float* hg    = hB + NF;                      // [64,32] pooled

    const int TILES = N_NODES / 16;              // 6250 (exact)
    const int WPB = 8;                           // 8 waves (256 threads) / block
    const dim3 gemmGrid((TILES + WPB - 1) / WPB);
    const dim3 gemmBlock(WPB * 32);
    const int edgeBlocks = (N_EDGES * 32) / 256;          // 400000
    const int reluBlocks = (int)((NF + 255) / 256);       // 12500
    const int poolBlocks = (N_NODES * 32 + 255) / 256;    // 12500

    // ---- Layer 0 (K = 128, input x) ----
    gemm_dual_wmma<<<gemmGrid, gemmBlock, 0, stream>>>(x, IN_FEATS, IN_FEATS,
                                                       Wn0, Ws0, bn0, rhs, neigh, N_NODES);
    edge_scatter<<<edgeBlocks, 256, 0, stream>>>(src, dst, rhs, neigh);
    relu_copy<<<reluBlocks, 256, 0, stream>>>(neigh, hA, (int)NF);

    // ---- Layers 1..3 (K = 32) ----
    float* hin = hA;
    float* hout = hB;
    for (int l = 0; l < 3; ++l) {
        gemm_dual_wmma<<<gemmGrid, gemmBlock, 0, stream>>>(hin, DIM, DIM,
                                                           Wn + (size_t)l * DIM * DIM,
                                                           Ws + (size_t)l * DIM * DIM,
                                                           bn + (size_t)l * DIM,
                                                           rhs, neigh, N_NODES);
        edge_scatter<<<edgeBlocks, 256, 0, stream>>>(src, dst, rhs, neigh);
        relu_copy<<<reluBlocks, 256, 0, stream>>>(neigh, hout, (int)NF);
        float* t = hin; hin = hout; hout = t;
    }

    // ---- Pool + head ----
    zero_buf<<<(N_GRAPHS * DIM + 255) / 256, 256, 0, stream>>>(hg, N_GRAPHS * DIM);
    pool_sum<<<poolBlocks, 256, 0, stream>>>(hin, gid, hg);
    head_kernel<<<1, 64, 0, stream>>>(hg, Wfc1, bfc1, Wout, bout, out);
}